// GraphConvolution_86457691668981
// MI455X (gfx1250) — compile-verified
//
#include <hip/hip_runtime.h>

typedef __attribute__((ext_vector_type(2))) float v2f;
typedef __attribute__((ext_vector_type(8))) float v8f;

#define GC_D      256
#define GC_N      100000
#define GC_NTILES (GC_N / 16)   // 6250 row tiles
#define LDSS      260           // padded LDS row stride in floats: 4*n mod 64 distinct -> conflict-free

// ---------------- zero the aggregation buffer ----------------
__global__ void gc_zero(float4* p, int n4) {
  int i = blockIdx.x * blockDim.x + threadIdx.x;
  if (i < n4) p[i] = make_float4(0.f, 0.f, 0.f, 0.f);
}

// ---------------- SpMM scatter: agg[dst] += w * feat[src] ----------------
// 64 threads per edge, float4 gathers, scalar f32 atomics (resolve in L2).
__global__ void gc_scatter(const float* __restrict__ feat,
                           const int*   __restrict__ src,
                           const int*   __restrict__ dst,
                           const float* __restrict__ wgt,
                           float*       __restrict__ agg, int E) {
  int t = blockIdx.x * blockDim.x + threadIdx.x;
  int e = t >> 6;
  if (e >= E) return;
  int c4 = (t & 63) << 2;
  int s = src[e], d = dst[e];
  float w = wgt[e];
  float4 f = *(const float4*)(feat + (size_t)s * GC_D + c4);
  float* o = agg + (size_t)d * GC_D + c4;
  atomicAdd(o + 0, w * f.x);
  atomicAdd(o + 1, w * f.y);
  atomicAdd(o + 2, w * f.z);
  atomicAdd(o + 3, w * f.w);
}

// ---------------- GEMM: out = agg @ W^T + b via V_WMMA_F32_16X16X4_F32 ----------------
// One wave owns one 16-row tile. A-panel (16x256) preloaded into 128 VGPRs in WMMA
// A layout. W staged into LDS in two 128-row phases (133 KB). In-place safe when
// agg == out because all global reads of the tile happen before any store.
__global__ __launch_bounds__(256) void gc_gemm(const float* __restrict__ agg,
                                               const float* __restrict__ Wm,
                                               const float* __restrict__ bias,
                                               float*       __restrict__ out) {
  __shared__ float lw[128 * LDSS];   // 133,120 bytes

  const int lane = threadIdx.x & 31;
  const int wave = threadIdx.x >> 5;
  const int hi   = (lane >> 4) & 1;  // 0: K%4 in {0,1}, 1: K%4 in {2,3}
  const int l15  = lane & 15;
  const int tile = blockIdx.x * 8 + wave;
  const bool valid = (tile < GC_NTILES);

  // Preload A tile: a[k] holds agg[row0 + l15][4k + 2*hi .. +1]  (ISA 16x4 F32 A layout)
  v2f a[64];
  if (valid) {
    const float* arow = agg + ((size_t)tile * 16 + l15) * GC_D;
#pragma unroll
    for (int k = 0; k < 64; ++k) {
      float4 v = *(const float4*)(arow + 4 * k);
      v2f av;
      av[0] = hi ? v.z : v.x;
      av[1] = hi ? v.w : v.y;
      a[k] = av;
    }
  }

  for (int h = 0; h < 2; ++h) {
    __syncthreads();  // protect LDS before overwrite (phase 2)
    // Stage W rows [h*128, h*128+128) into padded LDS.
    for (int i = threadIdx.x; i < 128 * 64; i += 256) {
      int r  = i >> 6;
      int k4 = (i & 63) << 2;
      *(float4*)&lw[r * LDSS + k4] =
          *(const float4*)(Wm + ((size_t)(h * 128 + r)) * GC_D + k4);
    }
    __syncthreads();

    if (valid) {
      for (int ct = 0; ct < 8; ++ct) {
        const int colg = h * 128 + ct * 16 + l15;  // global output column for this lane
        // B layout: VGPR0 = K rows {0,2}, VGPR1 = {1,3} -> lane reads W[colg][4k+2*hi..+1]
        const float* bb = &lw[(ct * 16 + l15) * LDSS + 2 * hi];
        v8f c = {};
#pragma unroll
        for (int k = 0; k < 64; ++k) {
          v2f bv = *(const v2f*)(bb + 4 * k);
          c = __builtin_amdgcn_wmma_f32_16x16x4_f32(false, a[k], false, bv,
                                                    (short)0, c, false, false);
        }
        float bsum = bias[colg];
        // C/D layout: VGPR r -> row r (lanes 0-15) / row r+8 (lanes 16-31)
        float* op = out + ((size_t)tile * 16 + hi * 8) * GC_D + colg;
#pragma unroll
        for (int r = 0; r < 8; ++r)
          op[(size_t)r * GC_D] = c[r] + bsum;
      }
    }
  }
}

// ---------------- host entry ----------------
extern "C" void kernel_launch(void* const* d_in, const int* in_sizes, int n_in,
                              void* d_out, int out_size, void* d_ws, size_t ws_size,
                              hipStream_t stream) {
  const float* feat = (const float*)d_in[0];
  const int*   esrc = (const int*)d_in[1];
  const int*   edst = (const int*)d_in[2];
  const float* ew   = (const float*)d_in[3];
  const float* Wm   = (const float*)d_in[4];
  const float* bias = (const float*)d_in[5];
  float* out = (float*)d_out;
  const int E = in_sizes[1];

  const size_t aggBytes = (size_t)GC_N * GC_D * sizeof(float);
  float* agg = (ws_size >= aggBytes) ? (float*)d_ws : out;  // deterministic per-run choice

  const int n4 = GC_N * GC_D / 4;
  gc_zero<<<(n4 + 255) / 256, 256, 0, stream>>>((float4*)agg, n4);

  const long long sthreads = (long long)E * 64;
  gc_scatter<<<(int)((sthreads + 255) / 256), 256, 0, stream>>>(feat, esrc, edst, ew, agg, E);

  const int blocks = (GC_NTILES + 7) / 8;  // one 16-row tile per wave, 8 waves/block
  gc_gemm<<<blocks, 256, 0, stream>>>(agg, Wm, bias, out);
}